// GATBase_3075196584517
// MI455X (gfx1250) — compile-verified
//
#include <hip/hip_runtime.h>
#include <math.h>

#define NN 50000
#define EE 600000
#define HH 4
#define FF 32
#define CDIM 128        // IN = H*F = 128
#define ETOT (EE + NN)  // edges + self loops
#define NEG_SLOPE 0.2f

typedef float v2f __attribute__((ext_vector_type(2)));
typedef float v8f __attribute__((ext_vector_type(8)));

// ---------------------------------------------------------------------------
// GEMM: h[N,128] = x[N,128] @ W[128,128] using V_WMMA_F32_16X16X4_F32.
// Block = 256 threads = 8 waves. Block b computes rows [16b,16b+16);
// wave w computes columns [16w,16w+16). 32 k-steps of 4.
// ---------------------------------------------------------------------------
__global__ void gat_gemm_wmma(const float* __restrict__ x,
                              const float* __restrict__ W,
                              float* __restrict__ h) {
    __shared__ float xs[16 * 132];   // padded stride 132 -> conflict-free column reads

    const int tid = threadIdx.x;
    const int rowbase = blockIdx.x * 16;

    // Stage the 16x128 x-tile in LDS (2048 floats / 256 threads = 8 each).
    for (int i = tid; i < 16 * CDIM; i += 256) {
        int r = i >> 7, c = i & 127;
        xs[r * 132 + c] = x[(size_t)(rowbase + r) * CDIM + c];
    }
    __syncthreads();

    const int wave = tid >> 5;
    const int lane = tid & 31;
    const int ln15 = lane & 15;
    const int half = lane >> 4;          // 0: K pair {0,1}; 1: K pair {2,3}
    const int colbase = wave * 16;

    v8f acc = {};
#pragma unroll
    for (int k0 = 0; k0 < CDIM; k0 += 4) {
        const int ka = k0 + 2 * half;
        v2f a, b;
        // A 16x4 f32: lane row = ln15, K = ka, ka+1
        a[0] = xs[ln15 * 132 + ka];
        a[1] = xs[ln15 * 132 + ka + 1];
        // B 4x16 f32: row = K, col = colbase + ln15 (W row-major [K][col])
        b[0] = W[(size_t)ka * CDIM + colbase + ln15];
        b[1] = W[(size_t)(ka + 1) * CDIM + colbase + ln15];
        acc = __builtin_amdgcn_wmma_f32_16x16x4_f32(
            /*neg_a=*/false, a, /*neg_b=*/false, b,
            /*c_mod=*/(short)0, acc, /*reuse_a=*/false, /*reuse_b=*/false);
    }

    // C/D layout: VGPR r -> M = r + 8*half, N = ln15
#pragma unroll
    for (int r = 0; r < 8; ++r) {
        int row = rowbase + r + 8 * half;
        h[(size_t)row * CDIM + colbase + ln15] = acc[r];
    }
}

// ---------------------------------------------------------------------------
// Per-node attention coefficients: a_s[n,h] = <h[n,h,:], att_src[h,:]>
// ---------------------------------------------------------------------------
__global__ void gat_attn_coeff(const float* __restrict__ h,
                               const float* __restrict__ att_src,
                               const float* __restrict__ att_dst,
                               float* __restrict__ a_s,
                               float* __restrict__ a_d) {
    int i = blockIdx.x * blockDim.x + threadIdx.x;   // over N*H
    if (i >= NN * HH) return;
    int node = i >> 2, hh = i & 3;
    const float* hp = h + (size_t)node * CDIM + hh * FF;
    const float* as = att_src + hh * FF;
    const float* ad = att_dst + hh * FF;
    float s = 0.f, d = 0.f;
#pragma unroll
    for (int f = 0; f < FF; ++f) {
        float v = hp[f];
        s = fmaf(v, as[f], s);
        d = fmaf(v, ad[f], d);
    }
    a_s[i] = s;
    a_d[i] = d;
}

// ---------------------------------------------------------------------------
// Init: out = 0, m = -inf, denom = 0
// ---------------------------------------------------------------------------
__global__ void gat_init(float* __restrict__ out, float* __restrict__ mx,
                         float* __restrict__ denom) {
    int i = blockIdx.x * blockDim.x + threadIdx.x;
    if (i < NN * CDIM) out[i] = 0.f;
    if (i < NN * HH) {
        mx[i] = -INFINITY;
        denom[i] = 0.f;
    }
}

// Exact float atomic max via signed/unsigned int ordering trick.
__device__ inline void atomicMaxFloat(float* addr, float val) {
    if (val >= 0.f)
        atomicMax((int*)addr, __float_as_int(val));
    else
        atomicMin((unsigned int*)addr, __float_as_uint(val));
}

__device__ inline void edge_endpoints(const int* __restrict__ ei, int e,
                                      int& src, int& dst) {
    if (e < EE) {
        src = ei[e];
        dst = ei[EE + e];
    } else {
        src = dst = e - EE;   // self loop
    }
}

// ---------------------------------------------------------------------------
// Edge pass 1: segment max of leaky_relu(a_s[src]+a_d[dst]) per head.
// ---------------------------------------------------------------------------
__global__ void gat_edge_max(const int* __restrict__ ei,
                             const float* __restrict__ a_s,
                             const float* __restrict__ a_d,
                             float* __restrict__ mx) {
    int e = blockIdx.x * blockDim.x + threadIdx.x;
    if (e >= ETOT) return;
    int src, dst;
    edge_endpoints(ei, e, src, dst);
    if ((unsigned)src >= NN || (unsigned)dst >= NN) return;
#pragma unroll
    for (int hh = 0; hh < HH; ++hh) {
        float v = a_s[src * HH + hh] + a_d[dst * HH + hh];
        v = (v > 0.f) ? v : NEG_SLOPE * v;
        atomicMaxFloat(&mx[dst * HH + hh], v);
    }
}

// ---------------------------------------------------------------------------
// Edge pass 2: denom[dst,h] += exp(e - m[dst,h])
// ---------------------------------------------------------------------------
__global__ void gat_edge_expsum(const int* __restrict__ ei,
                                const float* __restrict__ a_s,
                                const float* __restrict__ a_d,
                                const float* __restrict__ mx,
                                float* __restrict__ denom) {
    int e = blockIdx.x * blockDim.x + threadIdx.x;
    if (e >= ETOT) return;
    int src, dst;
    edge_endpoints(ei, e, src, dst);
    if ((unsigned)src >= NN || (unsigned)dst >= NN) return;
#pragma unroll
    for (int hh = 0; hh < HH; ++hh) {
        float v = a_s[src * HH + hh] + a_d[dst * HH + hh];
        v = (v > 0.f) ? v : NEG_SLOPE * v;
        float m = mx[dst * HH + hh];
        m = isfinite(m) ? m : 0.f;
        atomicAdd(&denom[dst * HH + hh], expf(v - m));
    }
}

// ---------------------------------------------------------------------------
// Edge pass 3: one wave32 per edge. lane = f index; 4 heads per lane.
// out[dst, h*32+f] += alpha[h] * hfeat[src, h*32+f]
// ---------------------------------------------------------------------------
__global__ void gat_edge_scatter(const int* __restrict__ ei,
                                 const float* __restrict__ a_s,
                                 const float* __restrict__ a_d,
                                 const float* __restrict__ mx,
                                 const float* __restrict__ denom,
                                 const float* __restrict__ hfeat,
                                 float* __restrict__ out) {
    int wave_in_blk = threadIdx.x >> 5;
    int lane = threadIdx.x & 31;
    int e = blockIdx.x * 8 + wave_in_blk;
    if (e >= ETOT) return;
    int src, dst;
    edge_endpoints(ei, e, src, dst);
    if ((unsigned)src >= NN || (unsigned)dst >= NN) return;

    float alpha[HH];
#pragma unroll
    for (int hh = 0; hh < HH; ++hh) {
        float v = a_s[src * HH + hh] + a_d[dst * HH + hh];
        v = (v > 0.f) ? v : NEG_SLOPE * v;
        float m = mx[dst * HH + hh];
        m = isfinite(m) ? m : 0.f;
        alpha[hh] = expf(v - m) / denom[dst * HH + hh];
    }
    const float* hp = hfeat + (size_t)src * CDIM;
    float* op = out + (size_t)dst * CDIM;
#pragma unroll
    for (int hh = 0; hh < HH; ++hh) {
        int c = hh * FF + lane;
        atomicAdd(&op[c], alpha[hh] * hp[c]);
    }
}

// ---------------------------------------------------------------------------
// Finalize: out = relu(out + bias)
// ---------------------------------------------------------------------------
__global__ void gat_finalize(float* __restrict__ out,
                             const float* __restrict__ bias) {
    int i = blockIdx.x * blockDim.x + threadIdx.x;
    if (i >= NN * CDIM) return;
    out[i] = fmaxf(out[i] + bias[i & 127], 0.f);
}

// ---------------------------------------------------------------------------
extern "C" void kernel_launch(void* const* d_in, const int* in_sizes, int n_in,
                              void* d_out, int out_size, void* d_ws, size_t ws_size,
                              hipStream_t stream) {
    const float* x       = (const float*)d_in[0];
    const int*   ei      = (const int*)d_in[1];
    const float* W       = (const float*)d_in[2];
    const float* att_src = (const float*)d_in[3];
    const float* att_dst = (const float*)d_in[4];
    const float* bias    = (const float*)d_in[5];
    float* out = (float*)d_out;

    // Workspace layout (floats): h[N*128] | a_s[N*4] | a_d[N*4] | m[N*4] | denom[N*4]
    float* ws    = (float*)d_ws;
    float* h     = ws;
    float* a_s   = h + (size_t)NN * CDIM;
    float* a_d   = a_s + (size_t)NN * HH;
    float* mx    = a_d + (size_t)NN * HH;
    float* denom = mx + (size_t)NN * HH;

    // 1) GEMM h = x @ W  (WMMA f32 16x16x4); 50000/16 = 3125 row tiles exactly
    gat_gemm_wmma<<<NN / 16, 256, 0, stream>>>(x, W, h);

    // 2) attention coefficients
    gat_attn_coeff<<<(NN * HH + 255) / 256, 256, 0, stream>>>(h, att_src, att_dst, a_s, a_d);

    // 3) init out / m / denom
    gat_init<<<(NN * CDIM + 255) / 256, 256, 0, stream>>>(out, mx, denom);

    // 4) edge passes
    gat_edge_max<<<(ETOT + 255) / 256, 256, 0, stream>>>(ei, a_s, a_d, mx);
    gat_edge_expsum<<<(ETOT + 255) / 256, 256, 0, stream>>>(ei, a_s, a_d, mx, denom);
    gat_edge_scatter<<<(ETOT + 7) / 8, 256, 0, stream>>>(ei, a_s, a_d, mx, denom, h, out);

    // 5) bias + relu
    gat_finalize<<<(NN * CDIM + 255) / 256, 256, 0, stream>>>(out, bias);
}